// SimplifiedHeteroGCN_7507602833967
// MI455X (gfx1250) — compile-verified
//
#include <hip/hip_runtime.h>
#include <hip/hip_bf16.h>

typedef __attribute__((ext_vector_type(2))) float v2f;
typedef __attribute__((ext_vector_type(8))) float v8f;

#define NN 100000
#define NE 1000000
#define DIM 64
#define NCLS 10

// ---------------------------------------------------------------------------
// f32 WMMA: D(16x16) = A(16x4) * B(4x16) + C   (V_WMMA_F32_16X16X4_F32)
// ---------------------------------------------------------------------------
__device__ __forceinline__ v8f wmma4(v2f a, v2f b, v8f c) {
  return __builtin_amdgcn_wmma_f32_16x16x4_f32(
      /*neg_a=*/false, a, /*neg_b=*/false, b,
      /*c_mod=*/(short)0, c, /*reuse_a=*/false, /*reuse_b=*/false);
}

// ---------------------------------------------------------------------------
// Utility kernels
// ---------------------------------------------------------------------------
__global__ void zero_f32(float* __restrict__ p, int n) {
  int i = blockIdx.x * blockDim.x + threadIdx.x;
  int stride = gridDim.x * blockDim.x;
  for (; i < n; i += stride) p[i] = 0.0f;
}

__global__ void degree_kernel(const int* __restrict__ dst, float* __restrict__ cnt, int e) {
  int i = blockIdx.x * blockDim.x + threadIdx.x;
  int stride = gridDim.x * blockDim.x;
  for (; i < e; i += stride) atomicAdd(&cnt[dst[i]], 1.0f);
}

__global__ void inv_kernel(float* __restrict__ cnt, int n) {
  int i = blockIdx.x * blockDim.x + threadIdx.x;
  int stride = gridDim.x * blockDim.x;
  for (; i < n; i += stride) cnt[i] = 1.0f / fmaxf(cnt[i], 1.0f);
}

// One thread per (edge, feature). Consecutive lanes share an edge and walk the
// 64-wide feature dim -> coalesced gather loads and coalesced f32 atomics,
// all L2-resident (node features = 25.6 MB << 192 MB L2).
__global__ void scatter_kernel(const float* __restrict__ feat,
                               const int* __restrict__ src,
                               const int* __restrict__ dst,
                               float* __restrict__ msg, int e) {
  int total = e * DIM;
  int i = blockIdx.x * blockDim.x + threadIdx.x;
  int stride = gridDim.x * blockDim.x;
  for (; i < total; i += stride) {
    int eidx = i >> 6;
    int f = i & 63;
    atomicAdd(&msg[(size_t)dst[eidx] * DIM + f],
              feat[(size_t)src[eidx] * DIM + f]);
  }
}

// ---------------------------------------------------------------------------
// SAGE layer: out[16 x 64] tile = (msg*inv) @ Wl^T + self @ Wr^T + b, ReLU
// One wave (32 lanes) per 16-node tile. K=64 -> 16 WMMA K-steps, two chained
// WMMAs per step per 16-col N-tile (mean branch + self branch) = 128 WMMAs.
// ---------------------------------------------------------------------------
__global__ void __launch_bounds__(32)
sage_wmma_kernel(const float* __restrict__ msg, const float* __restrict__ inv,
                 const float* __restrict__ self,
                 const float* __restrict__ Wl, const float* __restrict__ Wr,
                 const float* __restrict__ bias, float* __restrict__ out,
                 int relu) {
  const int lane = threadIdx.x;
  const int m = lane & 15;      // N index for B/C/D, M index for A
  const int half = lane >> 4;   // K-pair selector
  const int base = blockIdx.x * 16;
  const int node = base + m;

  const float2* mrow = (const float2*)(msg + (size_t)node * DIM);
  const float2* srow = (const float2*)(self + (size_t)node * DIM);
  const float sc = inv[node];

  v8f acc[4];
  acc[0] = (v8f){}; acc[1] = (v8f){}; acc[2] = (v8f){}; acc[3] = (v8f){};

#pragma unroll
  for (int k0 = 0; k0 < DIM; k0 += 4) {
    const int fi = (k0 >> 1) + half;      // float2 index: elements k0+2*half..+1
    float2 am2 = mrow[fi];
    float2 ax2 = srow[fi];
    v2f am; am.x = am2.x * sc; am.y = am2.y * sc;   // fused mean = msg * inv
    v2f ax; ax.x = ax2.x;      ax.y = ax2.y;
#pragma unroll
    for (int t = 0; t < 4; ++t) {
      const int n = t * 16 + m;           // output channel (row of W)
      float2 bl2 = ((const float2*)(Wl + (size_t)n * DIM))[fi];
      float2 br2 = ((const float2*)(Wr + (size_t)n * DIM))[fi];
      v2f bl; bl.x = bl2.x; bl.y = bl2.y;
      v2f br; br.x = br2.x; br.y = br2.y;
      acc[t] = wmma4(am, bl, acc[t]);
      acc[t] = wmma4(ax, br, acc[t]);
    }
  }

  // D layout: vgpr r, lane L -> row M = r + 8*(L/16), col N = L%16
#pragma unroll
  for (int t = 0; t < 4; ++t) {
    const float bv = bias[t * 16 + m];
#pragma unroll
    for (int r = 0; r < 8; ++r) {
      float v = acc[t][r] + bv;
      if (relu) v = fmaxf(v, 0.0f);
      out[(size_t)(base + r + 8 * half) * DIM + t * 16 + m] = v;
    }
  }
}

// ---------------------------------------------------------------------------
// Output head: out[16 x 10] tile = h @ W_out^T + b_out (single 16-col N-tile,
// only N<10 stored; weight row index clamped to stay in-bounds).
// ---------------------------------------------------------------------------
__global__ void __launch_bounds__(32)
head_wmma_kernel(const float* __restrict__ h, const float* __restrict__ W,
                 const float* __restrict__ bias, float* __restrict__ out) {
  const int lane = threadIdx.x;
  const int m = lane & 15;
  const int half = lane >> 4;
  const int base = blockIdx.x * 16;
  const int node = base + m;
  const int wn = (m < NCLS) ? m : 0;   // keep weight reads in-bounds

  const float2* hrow = (const float2*)(h + (size_t)node * DIM);
  const float2* wrow = (const float2*)(W + (size_t)wn * DIM);

  v8f acc = (v8f){};
#pragma unroll
  for (int k0 = 0; k0 < DIM; k0 += 4) {
    const int fi = (k0 >> 1) + half;
    float2 a2 = hrow[fi];
    float2 b2 = wrow[fi];
    v2f a; a.x = a2.x; a.y = a2.y;
    v2f b; b.x = b2.x; b.y = b2.y;
    acc = wmma4(a, b, acc);
  }

  if (m < NCLS) {
    const float bv = bias[m];
#pragma unroll
    for (int r = 0; r < 8; ++r) {
      out[(size_t)(base + r + 8 * half) * NCLS + m] = acc[r] + bv;
    }
  }
}

// ---------------------------------------------------------------------------
// Host-side orchestration
// ---------------------------------------------------------------------------
extern "C" void kernel_launch(void* const* d_in, const int* in_sizes, int n_in,
                              void* d_out, int out_size, void* d_ws, size_t ws_size,
                              hipStream_t stream) {
  (void)in_sizes; (void)n_in; (void)out_size; (void)ws_size;

  const float* x    = (const float*)d_in[0];
  const int*   ei   = (const int*)d_in[1];       // [2, E] row-major int32
  const float* W1l  = (const float*)d_in[2];
  const float* W1r  = (const float*)d_in[3];
  const float* b1   = (const float*)d_in[4];
  const float* W2l  = (const float*)d_in[5];
  const float* W2r  = (const float*)d_in[6];
  const float* b2   = (const float*)d_in[7];
  const float* Wout = (const float*)d_in[8];
  const float* bout = (const float*)d_in[9];
  float* out = (float*)d_out;

  const int* src = ei;
  const int* dst = ei + NE;

  // Workspace layout (floats): msg[N*64] | cnt/inv[N] | h1[N*64] | h2[N*64]
  float* ws  = (float*)d_ws;
  float* msg = ws;
  float* cnt = msg + (size_t)NN * DIM;
  float* h1  = cnt + NN;
  float* h2  = h1 + (size_t)NN * DIM;

  const int featTotal = NN * DIM;            // 6.4M
  const int scatTotal = NE * DIM;            // 64M items
  dim3 blk(256);
  dim3 gZeroBig((featTotal + NN + 255) / 256);
  dim3 gZero((featTotal + 255) / 256);
  dim3 gEdge((NE + 255) / 256);
  dim3 gNode((NN + 255) / 256);
  dim3 gScat((scatTotal + 255) / 256);
  dim3 gTile(NN / 16);                       // 6250 sixteen-node tiles
  dim3 wave(32);

  // Degree + inverse (computed once, reused by both layers)
  zero_f32<<<gZeroBig, blk, 0, stream>>>(msg, featTotal + NN);  // msg and cnt contiguous
  degree_kernel<<<gEdge, blk, 0, stream>>>(dst, cnt, NE);
  inv_kernel<<<gNode, blk, 0, stream>>>(cnt, NN);

  // Layer 1
  scatter_kernel<<<gScat, blk, 0, stream>>>(x, src, dst, msg, NE);
  sage_wmma_kernel<<<gTile, wave, 0, stream>>>(msg, cnt, x, W1l, W1r, b1, h1, 1);

  // Layer 2
  zero_f32<<<gZero, blk, 0, stream>>>(msg, featTotal);
  scatter_kernel<<<gScat, blk, 0, stream>>>(h1, src, dst, msg, NE);
  sage_wmma_kernel<<<gTile, wave, 0, stream>>>(msg, cnt, h1, W2l, W2r, b2, h2, 1);

  // Head
  head_wmma_kernel<<<gTile, wave, 0, stream>>>(h2, Wout, bout, out);
}